// TwoDWeights_LRPModel_32633161515023
// MI455X (gfx1250) — compile-verified
//
#include <hip/hip_runtime.h>

#define DD 4096

typedef __attribute__((ext_vector_type(2))) float v2f;
typedef __attribute__((ext_vector_type(8))) float v8f;

// ---------------------------------------------------------------------------
// Fused matvec + LRP scale vector:  t = W @ ain ;  act = relu?(t+b) ;
// scale = act / (t + 1e-9*sign(t)).  One wave32 per output row.
// ---------------------------------------------------------------------------
__global__ __launch_bounds__(256) void mv_lrp_kernel(
    const float* __restrict__ W, const float* __restrict__ ain,
    const float* __restrict__ b, float* __restrict__ act,
    float* __restrict__ scale, int relu)
{
  const int row  = (blockIdx.x * blockDim.x + threadIdx.x) >> 5;
  const int lane = threadIdx.x & 31;
  if (row >= DD) return;
  const float* wr = W + (size_t)row * DD;
  float sum = 0.f;
  for (int c = lane; c < DD; c += 32) sum += wr[c] * ain[c];
  #pragma unroll
  for (int off = 16; off > 0; off >>= 1) sum += __shfl_xor(sum, off, 32);
  if (lane == 0) {
    const float t   = sum;
    const float pre = t + b[row];
    const float a   = relu ? fmaxf(pre, 0.f) : pre;
    const float sg  = (t > 0.f) ? 1.f : ((t < 0.f) ? -1.f : 0.f);
    const float z   = t + 1e-9f * sg;
    act[row]   = a;
    scale[row] = a / z;
  }
}

// ---------------------------------------------------------------------------
// C = rowscale .* ( (A .* kscale_cols) @ B ) .* colscale      (4096^3 fp32)
// 128x128 block tile, BK=16, 8 waves, wave tile 32x64 (2x4 WMMA tiles).
// Double-buffered LDS fed by GLOBAL_LOAD_ASYNC_TO_LDS (ASYNCcnt pipeline):
//   s_wait_asynccnt 0 -> barrier -> issue next tile async -> WMMA on current.
// A stored [row][k] (80B rows: b128 async stores, b64 frag reads),
// B stored column-major [n][k] (72B rows: b64 frag reads, no reg shuffles).
// ---------------------------------------------------------------------------
template<bool HASK>
__global__ __launch_bounds__(256) void gemm_wmma_async(
    const float* __restrict__ A, const float* __restrict__ B,
    const float* __restrict__ kscale,
    const float* __restrict__ rowscale, const float* __restrict__ colscale,
    float* __restrict__ C)
{
  constexpr int BM = 128, BN = 128, BK = 16;
  constexpr int APAD = 20;   // floats/row -> 80B rows (16B aligned)
  constexpr int BPAD = 18;   // floats/row -> 72B rows (8B aligned)
  __shared__ float As[2][BM][APAD];
  __shared__ float Bs[2][BN][BPAD];

  const int tid   = threadIdx.x;
  const int bm    = blockIdx.y * BM;
  const int bn    = blockIdx.x * BN;
  const int wave  = tid >> 5;
  const int lane  = tid & 31;
  const int lhalf = lane >> 4;
  const int l16   = lane & 15;
  const int wm    = (wave & 3) * 32;   // 4 waves along M
  const int wn    = (wave >> 2) * 64;  // 2 waves along N

  const unsigned long long abase = (unsigned long long)(uintptr_t)A;
  const unsigned long long bbase = (unsigned long long)(uintptr_t)B;

  // Issue one BK-slab of async memory->LDS copies into buffer `buf`.
  auto issue_tile = [&](int k0, int buf) {
    // A tile: 128x16 floats = 512 float4; 2 x b128 per thread.
    #pragma unroll
    for (int i = 0; i < 2; ++i) {
      const int idx4 = tid + i * 256;
      const int r = idx4 >> 2, c4 = idx4 & 3;
      const unsigned goff = (unsigned)(((bm + r) * DD + k0 + c4 * 4) * 4);
      const unsigned loff = (unsigned)(uintptr_t)&As[buf][r][c4 * 4];
      asm volatile("global_load_async_to_lds_b128 %0, %1, %2"
                   :: "v"(loff), "v"(goff), "s"(abase) : "memory");
    }
    // B tile: 16x128 floats, transposed into Bs[n][k]; 8 x b32 per thread.
    #pragma unroll
    for (int i = 0; i < 8; ++i) {
      const int idx = tid + i * 256;
      const int r = idx >> 7, c = idx & 127;           // r = k, c = n
      const unsigned goff = (unsigned)(((k0 + r) * DD + bn + c) * 4);
      const unsigned loff = (unsigned)(uintptr_t)&Bs[buf][c][r];
      asm volatile("global_load_async_to_lds_b32 %0, %1, %2"
                   :: "v"(loff), "v"(goff), "s"(bbase) : "memory");
    }
  };

  v8f acc[2][4] = {};
  issue_tile(0, 0);

  for (int k0 = 0; k0 < DD; k0 += BK) {
    const int buf = (k0 >> 4) & 1;

    // My async copies for `buf` done; barrier => everyone's are, and everyone
    // has finished computing from buf^1, so buf^1 is safe to refill.
    asm volatile("s_wait_asynccnt 0x0" ::: "memory");
    __syncthreads();
    if (k0 + BK < DD) issue_tile(k0 + BK, buf ^ 1);

    v2f ksv[4];
    if (HASK) {
      #pragma unroll
      for (int s = 0; s < 4; ++s)
        ksv[s] = *(const v2f*)&kscale[k0 + s * 4 + lhalf * 2];
    }

    #pragma unroll
    for (int s = 0; s < 4; ++s) {          // K slabs of 4
      const int ka = s * 4 + lhalf * 2;    // K of VGPR0 for this lane
      v2f aF[2], bF[4];
      #pragma unroll
      for (int mi = 0; mi < 2; ++mi) {     // A frag: M=l16, K=ka+j  (b64)
        aF[mi] = *(const v2f*)&As[buf][wm + mi * 16 + l16][ka];
        if (HASK) { aF[mi][0] *= ksv[s][0]; aF[mi][1] *= ksv[s][1]; }
      }
      #pragma unroll
      for (int ni = 0; ni < 4; ++ni)       // B frag: K=ka+j, N=l16  (b64)
        bF[ni] = *(const v2f*)&Bs[buf][wn + ni * 16 + l16][ka];
      #pragma unroll
      for (int mi = 0; mi < 2; ++mi)
        #pragma unroll
        for (int ni = 0; ni < 4; ++ni)
          acc[mi][ni] = __builtin_amdgcn_wmma_f32_16x16x4_f32(
              false, aF[mi], false, bF[ni], (short)0, acc[mi][ni],
              false, false);
    }
  }

  // Epilogue: C/D layout -- VGPR r, lane l: M = (l>>4)*8 + r, N = l&15
  const bool doRow = (rowscale != nullptr);
  const bool doCol = (colscale != nullptr);
  #pragma unroll
  for (int mi = 0; mi < 2; ++mi) {
    #pragma unroll
    for (int ni = 0; ni < 4; ++ni) {
      const int gm = bm + wm + mi * 16;
      const int gn = bn + wn + ni * 16 + l16;
      const float cs = doCol ? colscale[gn] : 1.f;
      #pragma unroll
      for (int r = 0; r < 8; ++r) {
        const int row = gm + lhalf * 8 + r;
        float v = acc[mi][ni][r] * cs;
        if (doRow) v *= rowscale[row];
        C[(size_t)row * DD + gn] = v;
      }
    }
  }
}

// ---------------------------------------------------------------------------
// out = W3@relu(W2@relu(W1@x+b1)+b2)+b3
// R   = diag(out/z3) W3 diag(a2/z2) W2 diag(a1/z1) W1 diag(x)
//     = diag(r3) * [ ((W3 .* s2) @ W2) .* s1 ] @ W1 * diag(x)
// ---------------------------------------------------------------------------
extern "C" void kernel_launch(void* const* d_in, const int* in_sizes, int n_in,
                              void* d_out, int out_size, void* d_ws, size_t ws_size,
                              hipStream_t stream) {
  const float* x  = (const float*)d_in[0];
  const float* W1 = (const float*)d_in[1];
  const float* b1 = (const float*)d_in[2];
  const float* W2 = (const float*)d_in[3];
  const float* b2 = (const float*)d_in[4];
  const float* W3 = (const float*)d_in[5];
  const float* b3 = (const float*)d_in[6];

  float* out = (float*)d_out;                 // [D]
  float* R   = out + DD;                      // [D, D]

  float* C1 = (float*)d_ws;                   // [D, D] = ((W3 .* s2) @ W2) .* s1
  float* a1 = C1 + (size_t)DD * DD;
  float* s1 = a1 + DD;                        // a1/z1
  float* a2 = s1 + DD;
  float* s2 = a2 + DD;                        // a2/z2
  float* r3 = s2 + DD;                        // out/z3

  const dim3 mvGrid(DD / 8), mvBlk(256);      // 8 wave32 rows per block
  mv_lrp_kernel<<<mvGrid, mvBlk, 0, stream>>>(W1, x,  b1, a1,  s1, 1);
  mv_lrp_kernel<<<mvGrid, mvBlk, 0, stream>>>(W2, a1, b2, a2,  s2, 1);
  mv_lrp_kernel<<<mvGrid, mvBlk, 0, stream>>>(W3, a2, b3, out, r3, 0);

  const dim3 gGrid(DD / 128, DD / 128), gBlk(256);
  // C1 = ((W3 .* s2_cols) @ W2) .* s1_cols   (s1 folded into epilogue)
  gemm_wmma_async<true ><<<gGrid, gBlk, 0, stream>>>(W3, W2, s2, nullptr, s1, C1);
  // R  = diag(r3) @ (C1 @ W1) @ diag(x)      (pure A@B inner loop)
  gemm_wmma_async<false><<<gGrid, gBlk, 0, stream>>>(C1, W1, nullptr, r3, x, R);
}